// q_phi1_58506044506613
// MI455X (gfx1250) — compile-verified
//
#include <hip/hip_runtime.h>
#include <math.h>

typedef float v2f __attribute__((ext_vector_type(2)));
typedef float v8f __attribute__((ext_vector_type(8)));

#define NN   50000
#define EE   800000
#define FF   256
#define HH   256
#define CC   64
#define THETA 0.5f

// ---------------------------------------------------------------------------
// Degree / normalization
// ---------------------------------------------------------------------------
__global__ void k_init_deg(float* deg, int n) {
    int i = blockIdx.x * blockDim.x + threadIdx.x;
    if (i < n) deg[i] = 1.0f;                 // self loop
}

__global__ void k_count_deg(float* deg, const int* __restrict__ row, int e) {
    int i = blockIdx.x * blockDim.x + threadIdx.x;
    if (i < e) atomicAdd(&deg[row[i]], 1.0f);
}

__global__ void k_finalize_dinv(float* deg, int n) {
    int i = blockIdx.x * blockDim.x + threadIdx.x;
    if (i < n) {
        float d = deg[i];                      // always >= 1 here
        deg[i] = rsqrtf(d);
    }
}

__global__ void k_zero(float* a, int n) {
    int i = blockIdx.x * blockDim.x + threadIdx.x;
    if (i < n) a[i] = 0.0f;
}

// ---------------------------------------------------------------------------
// fp32 WMMA GEMM: C[M x N] = A[M x K] @ B[K x N] (+ bias[N] optional)
// One wave computes a 32x32 C tile (2x2 register blocking over 16x16 WMMA
// subtiles) -> each A/B fragment feeds two V_WMMA_F32_16X16X4_F32 ops.
// Fragment layout (ISA 7.12.2, 32-bit):
//   A 16x4 : lane(0-15) row=lane, K={k,k+1}; lane(16-31) row=lane-16, K={k+2,k+3}
//   B 4x16 : lane(0-15) col=lane, K rows {k,k+1}; lanes 16-31 cols, rows {k+2,k+3}
//   C 16x16: vgpr r, lane<16 -> (M=r,N=lane), lane>=16 -> (M=r+8,N=lane-16)
// M may end in a 16-row tail (M % 32 == 16): wave-uniform 1x2 variant.
// ---------------------------------------------------------------------------
__global__ void k_gemm_wmma_f32(const float* __restrict__ A,
                                const float* __restrict__ B,
                                float* __restrict__ C,
                                const float* __restrict__ bias,
                                int M, int N, int K,
                                int tilesN32, int totalTiles) {
    int wave = (int)((blockIdx.x * blockDim.x + threadIdx.x) >> 5);
    if (wave >= totalTiles) return;            // wave-uniform: EXEC stays all-1s
    int lane  = threadIdx.x & 31;
    int tm    = (wave / tilesN32) * 32;
    int tn    = (wave % tilesN32) * 32;
    int mrow  = lane & 15;
    int khalf = lane >> 4;                     // 0 or 1
    bool hasM1 = (tm + 16) < M;                // wave-uniform M-tail guard

    v8f acc00, acc01, acc10, acc11;
    float bv0 = bias ? bias[tn + mrow]      : 0.0f;
    float bv1 = bias ? bias[tn + 16 + mrow] : 0.0f;
#pragma unroll
    for (int r = 0; r < 8; ++r) {
        acc00[r] = bv0; acc01[r] = bv1;
        acc10[r] = bv0; acc11[r] = bv1;
    }

    const float* arow0 = A + (size_t)(tm + mrow) * K;
    const float* arow1 = arow0 + (size_t)16 * K;
    const float* bp0   = B + tn + mrow;
    const float* bp1   = bp0 + 16;

    if (hasM1) {
        for (int k = 0; k < K; k += 4) {
            int kk = k + khalf * 2;
            v2f a0, a1, b0, b1;
            a0.x = arow0[kk];  a0.y = arow0[kk + 1];
            a1.x = arow1[kk];  a1.y = arow1[kk + 1];
            b0.x = bp0[(size_t)kk * N];  b0.y = bp0[(size_t)(kk + 1) * N];
            b1.x = bp1[(size_t)kk * N];  b1.y = bp1[(size_t)(kk + 1) * N];
            acc00 = __builtin_amdgcn_wmma_f32_16x16x4_f32(false, a0, false, b0, (short)0, acc00, false, false);
            acc01 = __builtin_amdgcn_wmma_f32_16x16x4_f32(false, a0, false, b1, (short)0, acc01, false, false);
            acc10 = __builtin_amdgcn_wmma_f32_16x16x4_f32(false, a1, false, b0, (short)0, acc10, false, false);
            acc11 = __builtin_amdgcn_wmma_f32_16x16x4_f32(false, a1, false, b1, (short)0, acc11, false, false);
        }
    } else {
        for (int k = 0; k < K; k += 4) {
            int kk = k + khalf * 2;
            v2f a0, b0, b1;
            a0.x = arow0[kk];  a0.y = arow0[kk + 1];
            b0.x = bp0[(size_t)kk * N];  b0.y = bp0[(size_t)(kk + 1) * N];
            b1.x = bp1[(size_t)kk * N];  b1.y = bp1[(size_t)(kk + 1) * N];
            acc00 = __builtin_amdgcn_wmma_f32_16x16x4_f32(false, a0, false, b0, (short)0, acc00, false, false);
            acc01 = __builtin_amdgcn_wmma_f32_16x16x4_f32(false, a0, false, b1, (short)0, acc01, false, false);
        }
    }

    int mb = khalf * 8;
#pragma unroll
    for (int r = 0; r < 8; ++r) {
        C[(size_t)(tm + r + mb) * N + tn + mrow]      = acc00[r];
        C[(size_t)(tm + r + mb) * N + tn + 16 + mrow] = acc01[r];
    }
    if (hasM1) {
#pragma unroll
        for (int r = 0; r < 8; ++r) {
            C[(size_t)(tm + 16 + r + mb) * N + tn + mrow]      = acc10[r];
            C[(size_t)(tm + 16 + r + mb) * N + tn + 16 + mrow] = acc11[r];
        }
    }
}

// ---------------------------------------------------------------------------
// Aggregation: out[i] = dinv[i]^2 * h[i]  (self loop, also initializes out)
// then edge scatter: out[row] += dinv[row]*dinv[col]*h[col]  (f32 atomics, L2)
// Grid-stride over edges: each wave handles ~4 edges and prefetches the next
// edge's h[col] row (global_prefetch_b8) while the current atomics drain.
// ---------------------------------------------------------------------------
__global__ void k_self_init(const float* __restrict__ dinv,
                            const float* __restrict__ h,
                            float* __restrict__ out, int n4) {
    int i = blockIdx.x * blockDim.x + threadIdx.x;   // index in float4 units
    if (i >= n4) return;
    int node = i >> 6;                                // 64 float4 per row of 256
    float d = dinv[node];
    float w = d * d;
    float4 v = ((const float4*)h)[i];
    float4 o = make_float4(w * v.x, w * v.y, w * v.z, w * v.w);
    ((float4*)out)[i] = o;
}

__global__ void k_edge_scatter(const int* __restrict__ row,
                               const int* __restrict__ col,
                               const float* __restrict__ dinv,
                               const float* __restrict__ h,
                               float* out, int e) {
    int lane   = threadIdx.x & 31;
    int wave0  = (int)((blockIdx.x * blockDim.x + threadIdx.x) >> 5);
    int stride = (int)((gridDim.x * blockDim.x) >> 5);
    for (int ed = wave0; ed < e; ed += stride) {
        int en = ed + stride;
        if (en < e) {                               // prefetch next gather row
            int cn = col[en];
            __builtin_prefetch(h + (size_t)cn * FF + lane * 8, 0, 3);
        }
        int r = row[ed];
        int c = col[ed];
        float w = dinv[r] * dinv[c];
        const float4* hs = (const float4*)(h + (size_t)c * FF);
        float* ob = out + (size_t)r * FF;
#pragma unroll
        for (int half = 0; half < 2; ++half) {
            int i4 = lane + half * 32;
            float4 v = hs[i4];
            float* op = ob + i4 * 4;
            atomicAdd(op + 0, w * v.x);
            atomicAdd(op + 1, w * v.y);
            atomicAdd(op + 2, w * v.z);
            atomicAdd(op + 3, w * v.w);
        }
    }
}

__global__ void k_bias_act(float* __restrict__ x, const float* __restrict__ b,
                           int n4, int do_relu) {
    int i = blockIdx.x * blockDim.x + threadIdx.x;
    if (i >= n4) return;
    float4 bb = ((const float4*)b)[i & 63];
    float4 v  = ((float4*)x)[i];
    v.x += bb.x; v.y += bb.y; v.z += bb.z; v.w += bb.w;
    if (do_relu) {
        v.x = fmaxf(v.x, 0.f); v.y = fmaxf(v.y, 0.f);
        v.z = fmaxf(v.z, 0.f); v.w = fmaxf(v.w, 0.f);
    }
    ((float4*)x)[i] = v;
}

// ---------------------------------------------------------------------------
// Double L2 normalize, one wave (32 lanes) per node of 256 floats
// ---------------------------------------------------------------------------
__global__ void k_l2norm2(float* __restrict__ x, int n) {
    int wave = (int)((blockIdx.x * blockDim.x + threadIdx.x) >> 5);
    if (wave >= n) return;
    int lane = threadIdx.x & 31;
    float4* p = (float4*)(x + (size_t)wave * FF);
    float4 v0 = p[lane];
    float4 v1 = p[lane + 32];
    float s = v0.x * v0.x + v0.y * v0.y + v0.z * v0.z + v0.w * v0.w
            + v1.x * v1.x + v1.y * v1.y + v1.z * v1.z + v1.w * v1.w;
#pragma unroll
    for (int off = 16; off > 0; off >>= 1) s += __shfl_xor(s, off, 32);
    float n1 = sqrtf(s);
    float s1 = 1.0f / fmaxf(n1, 1e-12f);
    float n2 = n1 * s1;                         // norm after first normalize
    float sc = s1 * (1.0f / fmaxf(n2, 1e-12f)); // combined double-normalize scale
    v0.x *= sc; v0.y *= sc; v0.z *= sc; v0.w *= sc;
    v1.x *= sc; v1.y *= sc; v1.z *= sc; v1.w *= sc;
    p[lane]      = v0;
    p[lane + 32] = v1;
}

// ---------------------------------------------------------------------------
// Reconstruction loss: one wave per edge id; 3 dot products of 256;
// block-level LDS reduction, 4 global atomics per block.
// accum: [0]=neg_loss [1]=pos_loss [2]=n_neg [3]=n_pos
// ---------------------------------------------------------------------------
__global__ void k_edge_loss(const int* __restrict__ e0, const int* __restrict__ e1,
                            const int* __restrict__ g0, const int* __restrict__ g1,
                            const float* __restrict__ ls,
                            const float* __restrict__ rep,
                            const float* __restrict__ feat,
                            float* accum, int e) {
    __shared__ float part[4];
    if (threadIdx.x < 4) part[threadIdx.x] = 0.0f;
    __syncthreads();

    int wave = (int)((blockIdx.x * blockDim.x + threadIdx.x) >> 5);
    int lane = threadIdx.x & 31;
    if (wave < e) {
        int a = e0[wave], b = e1[wave];
        int c = g0[wave], d = g1[wave];
        const float4* ra = (const float4*)(rep  + (size_t)a * FF);
        const float4* rb = (const float4*)(rep  + (size_t)b * FF);
        const float4* rc = (const float4*)(rep  + (size_t)c * FF);
        const float4* rd = (const float4*)(rep  + (size_t)d * FF);
        const float4* fa = (const float4*)(feat + (size_t)a * FF);
        const float4* fb = (const float4*)(feat + (size_t)b * FF);
        float dp = 0.f, df = 0.f, dn = 0.f;
#pragma unroll
        for (int half = 0; half < 2; ++half) {
            int i4 = lane + half * 32;
            float4 x0 = ra[i4], x1 = rb[i4];
            dp += x0.x * x1.x + x0.y * x1.y + x0.z * x1.z + x0.w * x1.w;
            float4 f0 = fa[i4], f1 = fb[i4];
            df += f0.x * f1.x + f0.y * f1.y + f0.z * f1.z + f0.w * f1.w;
            float4 y0 = rc[i4], y1 = rd[i4];
            dn += y0.x * y1.x + y0.y * y1.y + y0.z * y1.z + y0.w * y1.w;
        }
#pragma unroll
        for (int off = 16; off > 0; off >>= 1) {
            dp += __shfl_xor(dp, off, 32);
            df += __shfl_xor(df, off, 32);
            dn += __shfl_xor(dn, off, 32);
        }
        if (lane == 0) {
            float pm = (a < b) ? 1.0f : 0.0f;
            float nm = (c < d) ? 1.0f : 0.0f;
            float nw = fmaxf(dn, 0.0f);
            float pw = fmaxf(dp, 0.0f);
            float pos  = df * THETA + pw * (1.0f - THETA);
            float diff = pos - ls[wave];
            atomicAdd(&part[0], nm * nw * nw);
            atomicAdd(&part[1], pm * diff * diff);
            atomicAdd(&part[2], nm);
            atomicAdd(&part[3], pm);
        }
    }
    __syncthreads();
    if (threadIdx.x < 4) atomicAdd(&accum[threadIdx.x], part[threadIdx.x]);
}

__global__ void k_final_loss(const float* __restrict__ accum, float* out1) {
    if (blockIdx.x == 0 && threadIdx.x == 0)
        out1[0] = (accum[0] + accum[1]) * (float)NN / (accum[2] + accum[3]);
}

// ---------------------------------------------------------------------------
// Launch
// ---------------------------------------------------------------------------
extern "C" void kernel_launch(void* const* d_in, const int* in_sizes, int n_in,
                              void* d_out, int out_size, void* d_ws, size_t ws_size,
                              hipStream_t stream) {
    const int*   ei   = (const int*)d_in[0];     // [2,E]
    const float* feat = (const float*)d_in[1];   // [N,256]
    const float* ls   = (const float*)d_in[2];   // [E]
    const int*   nei  = (const int*)d_in[3];     // [2,E]
    const float* W1   = (const float*)d_in[4];
    const float* b1   = (const float*)d_in[5];
    const float* W2   = (const float*)d_in[6];
    const float* b2   = (const float*)d_in[7];
    const float* Wy   = (const float*)d_in[8];
    const float* by   = (const float*)d_in[9];

    const int* row = ei;           // edge_index[0]
    const int* col = ei + EE;      // edge_index[1]
    const int* n0  = nei;
    const int* n1  = nei + EE;

    float* ws   = (float*)d_ws;
    float* dinv = ws;                                  // N floats
    float* hbuf = ws + (1 << 16);                      // N*256 floats (staging)
    float* acc  = hbuf + (size_t)NN * FF;              // 8 floats

    float* outp = (float*)d_out;
    float* rep  = outp;                                // N*256 (also x1/x2 scratch)
    float* rec  = outp + (size_t)NN * FF;              // 1
    float* y    = rec + 1;                             // N*64 (4B aligned)

    const int T = 256;
    int nb_n   = (NN + T - 1) / T;
    int nb_e   = (EE + T - 1) / T;
    int nb_ews = (EE / 4 + 7) / 8;                     // ~4 edges per wave, 8 waves/blk
    int nb_el4 = ((NN * (FF / 4)) + T - 1) / T;        // N*64 float4s
    int nb_nw  = (NN + 7) / 8;                         // wave per node
    int nb_ew  = (EE + 7) / 8;                         // wave per edge (loss)

    // degrees -> dinv
    k_init_deg<<<nb_n, T, 0, stream>>>(dinv, NN);
    k_count_deg<<<nb_e, T, 0, stream>>>(dinv, row, EE);
    k_finalize_dinv<<<nb_n, T, 0, stream>>>(dinv, NN);
    k_zero<<<1, 8, 0, stream>>>(acc, 8);

    int tilesM32 = (NN + 31) / 32;                     // 1563 (last tile is 16 rows)

    // conv1: h1 = feat @ W1 ; x1 = agg(h1) + b1 ; relu   (x1 lives in rep region)
    {
        int tilesN32 = HH / 32, total = tilesM32 * tilesN32;
        k_gemm_wmma_f32<<<(total + 7) / 8, T, 0, stream>>>(
            feat, W1, hbuf, nullptr, NN, HH, FF, tilesN32, total);
    }
    k_self_init<<<nb_el4, T, 0, stream>>>(dinv, hbuf, rep, NN * (FF / 4));
    k_edge_scatter<<<nb_ews, T, 0, stream>>>(row, col, dinv, hbuf, rep, EE);
    k_bias_act<<<nb_el4, T, 0, stream>>>(rep, b1, NN * (FF / 4), 1);

    // conv2: h2 = x1 @ W2 ; x2 = agg(h2) + b2    (x2 overwrites x1 in rep region)
    {
        int tilesN32 = HH / 32, total = tilesM32 * tilesN32;
        k_gemm_wmma_f32<<<(total + 7) / 8, T, 0, stream>>>(
            rep, W2, hbuf, nullptr, NN, HH, HH, tilesN32, total);
    }
    k_self_init<<<nb_el4, T, 0, stream>>>(dinv, hbuf, rep, NN * (FF / 4));
    k_edge_scatter<<<nb_ews, T, 0, stream>>>(row, col, dinv, hbuf, rep, EE);
    k_bias_act<<<nb_el4, T, 0, stream>>>(rep, b2, NN * (FF / 4), 0);

    // rep = l2norm(l2norm(x2)) in place
    k_l2norm2<<<nb_nw, T, 0, stream>>>(rep, NN);

    // y = rep @ Wy + by
    {
        int tilesN32 = CC / 32, total = tilesM32 * tilesN32;
        k_gemm_wmma_f32<<<(total + 7) / 8, T, 0, stream>>>(
            rep, Wy, y, by, NN, CC, HH, tilesN32, total);
    }

    // reconstruction loss
    k_edge_loss<<<nb_ew, T, 0, stream>>>(row, col, n0, n1, ls, rep, feat, acc, EE);
    k_final_loss<<<1, 1, 0, stream>>>(acc, rec);
}